// Net_80977313399076
// MI455X (gfx1250) — compile-verified
//
#include <hip/hip_runtime.h>
#include <hip/hip_bf16.h>

// ---------------------------------------------------------------------------
// CDNA5 (gfx1250) GCN link-prediction pipeline.
// Dense GEMMs use V_WMMA_F32_16X16X4_F32 (fp32 A/B, fp32 accum) — exact
// w.r.t. the fp32 reference. K and N are compile-time template constants so
// the K-loop fully unrolls and all B loads use immediate global offsets.
// Edge aggregation: wave-per-edge coalesced gathers + global_atomic_add_f32.
// ---------------------------------------------------------------------------

typedef float v2f __attribute__((ext_vector_type(2)));
typedef float v8f __attribute__((ext_vector_type(8)));

// C = A[M,K] @ B[K,N], N == 128, M % 16 == 0. Block: 256 threads (8 waves).
// Wave w computes the 16x16 tile at columns [16w, 16w+16).
// Optional ReLU applied to A while staging into LDS.
template <int K, int RELU_A>
__global__ __launch_bounds__(256) void gemm16_wmma_f32(
    const float* __restrict__ A, const float* __restrict__ B,
    float* __restrict__ C)
{
    constexpr int N = 128;
    constexpr int LDS_STRIDE = K + 4;   // pad: lane L hits banks 4*L apart
    __shared__ float As[16 * LDS_STRIDE];

    const int m0  = blockIdx.x * 16;
    const int tid = threadIdx.x;

    // B is small (<=128 KB) and re-read by every block: keep it warm.
    __builtin_prefetch(B + tid, 0, 3);

    // Cooperative stage of the 16 x K A-strip (coalesced along K).
#pragma unroll
    for (int idx = tid; idx < 16 * K; idx += 256) {
        const int r = idx / K;
        const int c = idx - r * K;
        float v = A[(size_t)(m0 + r) * K + c];
        if (RELU_A) v = fmaxf(v, 0.0f);
        As[r * LDS_STRIDE + c] = v;
    }
    __syncthreads();

    const int wave = tid >> 5;
    const int lane = tid & 31;
    const int nn   = wave * 16 + (lane & 15);  // output column of this lane
    const int kg   = (lane >> 4) * 2;          // lanes 0-15: K {k,k+1}; 16-31: {k+2,k+3}
    const int row  = lane & 15;                // A row within the strip

    v8f acc = {};
    const float* Arow = &As[row * LDS_STRIDE + kg];
    const float* Bl   = B + (size_t)kg * N + nn;   // per-lane base; rest is immediate

#pragma unroll
    for (int k = 0; k < K; k += 4) {
        v2f a, b;
        // 16x4 fp32 A fragment: VGPR0 = K=k+kg, VGPR1 = K=k+kg+1
        a.x = Arow[k];
        a.y = Arow[k + 1];
        // 4x16 fp32 B fragment: VGPR0 lanes0-15 K=k (16-31: k+2); VGPR1 K=k+1 (k+3)
        b.x = Bl[(size_t)k * N];
        b.y = Bl[(size_t)(k + 1) * N];
        // (neg_a, A, neg_b, B, c_mod, C, reuse_a, reuse_b)
        acc = __builtin_amdgcn_wmma_f32_16x16x4_f32(
            false, a, false, b, (short)0, acc, false, false);
    }

    // 16x16 f32 D layout: VGPR v -> row v (lanes 0-15) / row v+8 (lanes 16-31)
    float* Cl = C + (size_t)(m0 + (lane >> 4) * 8) * N + nn;
#pragma unroll
    for (int v = 0; v < 8; ++v) {
        Cl[(size_t)v * N] = acc[v];
    }
}

__global__ __launch_bounds__(256) void zero_f32(float* __restrict__ p, int n)
{
    const int i = blockIdx.x * 256 + threadIdx.x;
    if (i < n) p[i] = 0.0f;
}

// agg[dst[e]][0:128] += t[src[e]][0:128]; one wave32 per edge.
__global__ __launch_bounds__(256) void scatter_add128(
    const float* __restrict__ t, const int* __restrict__ src,
    const int* __restrict__ dst, float* __restrict__ agg, int E)
{
    const int w    = (int)(((size_t)blockIdx.x * 256 + threadIdx.x) >> 5);
    const int lane = threadIdx.x & 31;
    if (w >= E) return;
    const int s = src[w];
    const int d = dst[w];
    const float4 v = *(const float4*)(t + (size_t)s * 128 + lane * 4);
    float* o = agg + (size_t)d * 128 + lane * 4;
    __hip_atomic_fetch_add(o + 0, v.x, __ATOMIC_RELAXED, __HIP_MEMORY_SCOPE_AGENT);
    __hip_atomic_fetch_add(o + 1, v.y, __ATOMIC_RELAXED, __HIP_MEMORY_SCOPE_AGENT);
    __hip_atomic_fetch_add(o + 2, v.z, __ATOMIC_RELAXED, __HIP_MEMORY_SCOPE_AGENT);
    __hip_atomic_fetch_add(o + 3, v.w, __ATOMIC_RELAXED, __HIP_MEMORY_SCOPE_AGENT);
}

// out[e, 0:2] = [z[a]||z[b]] @ Wlin, one wave32 per edge pair.
__global__ __launch_bounds__(256) void decode_pairs(
    const float* __restrict__ z,
    const int* __restrict__ pos0, const int* __restrict__ pos1,
    const int* __restrict__ neg0, const int* __restrict__ neg1,
    const float* __restrict__ Wlin, float* __restrict__ out, int NP)
{
    const int w    = (int)(((size_t)blockIdx.x * 256 + threadIdx.x) >> 5);
    const int lane = threadIdx.x & 31;
    if (w >= 2 * NP) return;

    __builtin_prefetch(Wlin + lane * 16, 0, 3);   // 512 floats, L2-resident

    int a, b;
    if (w < NP) { a = pos0[w];      b = pos1[w];      }
    else        { a = neg0[w - NP]; b = neg1[w - NP]; }

    const float4 za = *(const float4*)(z + (size_t)a * 128 + lane * 4);
    const float4 zb = *(const float4*)(z + (size_t)b * 128 + lane * 4);

    const int j = lane * 4;
    float p0 = 0.0f, p1 = 0.0f;
    const float* va = (const float*)&za;
    const float* vb = (const float*)&zb;
#pragma unroll
    for (int u = 0; u < 4; ++u) {
        p0 += va[u] * Wlin[(j + u) * 2 + 0] + vb[u] * Wlin[(128 + j + u) * 2 + 0];
        p1 += va[u] * Wlin[(j + u) * 2 + 1] + vb[u] * Wlin[(128 + j + u) * 2 + 1];
    }
#pragma unroll
    for (int o = 16; o > 0; o >>= 1) {
        p0 += __shfl_xor(p0, o, 32);
        p1 += __shfl_xor(p1, o, 32);
    }
    if (lane == 0) {
        out[(size_t)w * 2 + 0] = p0;
        out[(size_t)w * 2 + 1] = p1;
    }
}

extern "C" void kernel_launch(void* const* d_in, const int* in_sizes, int n_in,
                              void* d_out, int out_size, void* d_ws, size_t ws_size,
                              hipStream_t stream)
{
    (void)n_in; (void)out_size; (void)ws_size;

    const float* x    = (const float*)d_in[0];   // [50000, 256]
    const int*   ei   = (const int*)d_in[1];     // [2, 500000]
    const int*   pei  = (const int*)d_in[2];     // [2, 100000]
    const int*   nei  = (const int*)d_in[3];     // [2, 100000]
    const float* W1   = (const float*)d_in[4];   // [256, 128]
    const float* W2   = (const float*)d_in[5];   // [128, 128]
    const float* Wlin = (const float*)d_in[6];   // [256, 2]
    float*       out  = (float*)d_out;           // [200000, 2]

    const int N_NODES = 50000;
    const int HID     = 128;
    const int E       = in_sizes[1] / 2;         // 500000
    const int NP      = in_sizes[2] / 2;         // 100000

    float* buf0 = (float*)d_ws;                  // t1 / t2    : 50000*128
    float* buf1 = buf0 + (size_t)N_NODES * HID;  // agg1 / z   : 50000*128
    const int NODE_ELEMS = N_NODES * HID;        // 6.4M

    const int* src = ei;
    const int* dst = ei + E;

    // t1 = x @ W1
    gemm16_wmma_f32<256, 0><<<N_NODES / 16, 256, 0, stream>>>(x, W1, buf0);
    // agg1 = segment_sum(t1[src], dst)
    zero_f32<<<(NODE_ELEMS + 255) / 256, 256, 0, stream>>>(buf1, NODE_ELEMS);
    scatter_add128<<<(E * 32 + 255) / 256, 256, 0, stream>>>(buf0, src, dst, buf1, E);
    // t2 = relu(agg1) @ W2
    gemm16_wmma_f32<128, 1><<<N_NODES / 16, 256, 0, stream>>>(buf1, W2, buf0);
    // z = segment_sum(t2[src], dst)
    zero_f32<<<(NODE_ELEMS + 255) / 256, 256, 0, stream>>>(buf1, NODE_ELEMS);
    scatter_add128<<<(E * 32 + 255) / 256, 256, 0, stream>>>(buf0, src, dst, buf1, E);
    // out = [z[ei0] || z[ei1]] @ Wlin
    decode_pairs<<<(2 * NP * 32 + 255) / 256, 256, 0, stream>>>(
        buf1, pei, pei + NP, nei, nei + NP, Wlin, out, NP);
}